// SpectralConv2d_65592740545049
// MI455X (gfx1250) — compile-verified
//
#include <hip/hip_runtime.h>

// MI455X / gfx1250 implementation of the DHT spectral conv.
// All transforms expressed as two-stage GEMMs on V_WMMA_F32_16X16X4_F32.

typedef __attribute__((ext_vector_type(2))) float v2f;
typedef __attribute__((ext_vector_type(8))) float v8f;

__device__ __forceinline__ v8f wmma4(v2f a, v2f b, v8f c) {
  return __builtin_amdgcn_wmma_f32_16x16x4_f32(false, a, false, b, (short)0, c,
                                               false, false);
}
__device__ __forceinline__ v8f zero8() {
  v8f c = {0.f, 0.f, 0.f, 0.f, 0.f, 0.f, 0.f, 0.f};
  return c;
}

#define NIMG 1024   // B*C
#define PIX  65536  // 256*256
#define CORN 1024   // 32*32
#define TWO_PI_256 0.0245436926061702596f  // 2*pi/256

// ---------------- transform-matrix init (integer-mod phase => exact) -------
// M1a  [128][256] : rows k*32+u      (k: C,S,S1,C1), cols m
// W2a  [32][1024] : rows v, cols k*256+n, signs (+C, -S, -C1, -S1)
// M1s  [1024][32] : rows k*256+U, cols u
// W2s  [256][128] : rows V, cols k*32+v, signs (+C, -S, -C1, -S1)
__global__ void init_mats(float* M1a, float* W2a, float* M1s, float* W2s) {
  int t = blockIdx.x * blockDim.x + threadIdx.x;
  if (t < 32768) {
    int r = t >> 8, m = t & 255, k = r >> 5, u = r & 31;
    float a0 = ((u * m) & 255) * TWO_PI_256;
    float a1 = ((u * (m + 1)) & 255) * TWO_PI_256;
    M1a[t] = (k == 0) ? __cosf(a0) : (k == 1) ? __sinf(a0)
           : (k == 2) ? __sinf(a1) : __cosf(a1);
  } else if (t < 65536) {
    int q = t - 32768;
    int v = q >> 10, j = q & 1023, k = j >> 8, n = j & 255;
    float a0 = ((v * n) & 255) * TWO_PI_256;
    float a1 = ((v * (n + 1)) & 255) * TWO_PI_256;
    W2a[q] = (k == 0) ? __cosf(a0) : (k == 1) ? -__sinf(a0)
           : (k == 2) ? -__cosf(a1) : -__sinf(a1);
  } else if (t < 98304) {
    int q = t - 65536;
    int r = q >> 5, u = q & 31, k = r >> 8, U = r & 255;
    float a0 = ((U * u) & 255) * TWO_PI_256;
    float a1 = ((U * (u + 1)) & 255) * TWO_PI_256;
    M1s[q] = (k == 0) ? __cosf(a0) : (k == 1) ? __sinf(a0)
           : (k == 2) ? __sinf(a1) : __cosf(a1);
  } else if (t < 131072) {
    int q = t - 98304;
    int V = q >> 7, j = q & 127, k = j >> 5, v = j & 31;
    float a0 = ((V * v) & 255) * TWO_PI_256;
    float a1 = ((V * (v + 1)) & 255) * TWO_PI_256;
    W2s[q] = (k == 0) ? __cosf(a0) : (k == 1) ? -__sinf(a0)
           : (k == 2) ? -__cosf(a1) : -__sinf(a1);
  }
}

// ---------------- analysis: 32x32 spectral corner of one image -------------
// Stage1: Z[128][256] = M1a * X  (X optionally masked by |x-xrec|>0.1)
// Stage2: corner[32][32] = sum_k Z_k * (sign * M2_k)^T   (K = 4*256 = 1024)
template <bool REFINE>
__global__ void dht_analysis(const float* __restrict__ x,
                             const float* __restrict__ xrec,
                             const float* __restrict__ M1a,
                             const float* __restrict__ W2a,
                             float* __restrict__ out_corner) {
  extern __shared__ float Z[];  // 128*256 floats = 128 KB
  const int img = blockIdx.x;
  const int tid = threadIdx.x;
  const int wave = tid >> 5, lane = tid & 31;
  const int lo = lane & 15, hi = lane >> 4;
  const float* X = x + (size_t)img * PIX;
  const float* XR = REFINE ? (xrec + (size_t)img * PIX) : nullptr;

  for (int t = wave; t < 128; t += 8) {          // 8x16 tiles of 16x16
    const int mt = t & 7, nb = (t >> 3) * 16;
    v8f c = zero8();
    const int arow = (mt * 16 + lo) * 256;
    for (int kk = 0; kk < 256; kk += 4) {
      v2f a, b;
      const int ac = kk + 2 * hi;
      a.x = M1a[arow + ac];
      a.y = M1a[arow + ac + 1];
      const int br = ac * 256 + nb + lo;
      float bx = X[br], by = X[br + 256];
      if (REFINE) {
        const float rx = XR[br], ry = XR[br + 256];
        bx = (__builtin_fabsf(bx - rx) > 0.1f) ? bx : 0.0f;
        by = (__builtin_fabsf(by - ry) > 0.1f) ? by : 0.0f;
      }
      b.x = bx; b.y = by;
      c = wmma4(a, b, c);
    }
    const int zr = mt * 16 + 8 * hi, zc = nb + lo;
#pragma unroll
    for (int r = 0; r < 8; ++r) Z[(zr + r) * 256 + zc] = c[r];
  }
  __syncthreads();

  if (wave < 4) {                                 // 2x2 output tiles
    const int ut = wave >> 1, vt = wave & 1;
    v8f c = zero8();
    for (int s = 0; s < 256; ++s) {
      const int kblk = s >> 6, nb = (s & 63) * 4;
      v2f a, b;
      const int ar = (kblk * 32 + ut * 16 + lo) * 256 + nb + 2 * hi;
      a.x = Z[ar]; a.y = Z[ar + 1];
      const int br = (vt * 16 + lo) * 1024 + kblk * 256 + nb + 2 * hi;
      b.x = W2a[br]; b.y = W2a[br + 1];
      c = wmma4(a, b, c);
    }
    float* O = out_corner + (size_t)img * CORN;
#pragma unroll
    for (int r = 0; r < 8; ++r)
      O[(ut * 16 + r + 8 * hi) * 32 + vt * 16 + lo] = c[r];
  }
}

// ---------------- synthesis: full 256x256 from zero-padded 32x32 corner ----
// Stage1: Zs[1024][32] = M1s * O ; Stage2: Y = sum_k Zs_k * (sign M2_k)^T, K=128
__global__ void dht_synthesis(const float* __restrict__ corner,
                              const float* __restrict__ M1s,
                              const float* __restrict__ W2s,
                              float* __restrict__ out, float scale) {
  extern __shared__ float Zs[];  // 1024*32 floats = 128 KB
  const int img = blockIdx.x;
  const int tid = threadIdx.x;
  const int wave = tid >> 5, lane = tid & 31;
  const int lo = lane & 15, hi = lane >> 4;
  const float* O = corner + (size_t)img * CORN;

  for (int t = wave; t < 128; t += 8) {           // 64x2 tiles
    const int mt = t & 63, nb = (t >> 6) * 16;
    v8f c = zero8();
    const int arow = (mt * 16 + lo) * 32;
    for (int kk = 0; kk < 32; kk += 4) {
      v2f a, b;
      const int ac = kk + 2 * hi;
      a.x = M1s[arow + ac];
      a.y = M1s[arow + ac + 1];
      const int br = ac * 32 + nb + lo;
      b.x = O[br]; b.y = O[br + 32];
      c = wmma4(a, b, c);
    }
    const int zr = mt * 16 + 8 * hi, zc = nb + lo;
#pragma unroll
    for (int r = 0; r < 8; ++r) Zs[(zr + r) * 32 + zc] = c[r];
  }
  __syncthreads();

  float* Y = out + (size_t)img * PIX;
  for (int t = wave; t < 256; t += 8) {           // 16x16 output tiles
    const int Ut = t & 15, Vt = t >> 4;
    v8f c = zero8();
    for (int s = 0; s < 32; ++s) {
      const int kblk = s >> 3, vb = (s & 7) * 4;
      v2f a, b;
      const int ar = (kblk * 256 + Ut * 16 + lo) * 32 + vb + 2 * hi;
      a.x = Zs[ar]; a.y = Zs[ar + 1];
      const int br = (Vt * 16 + lo) * 128 + kblk * 32 + vb + 2 * hi;
      b.x = W2s[br]; b.y = W2s[br + 1];
      c = wmma4(a, b, c);
    }
#pragma unroll
    for (int r = 0; r < 8; ++r)
      Y[(Ut * 16 + r + 8 * hi) * 256 + Vt * 16 + lo] = c[r] * scale;
  }
}

// ---------------- channel mix: out[b,o] (+)= sum_i in[b,i] * w[i,o] @ pos p --
__global__ void chan_mix(const float* __restrict__ in_c,   // [16][64][1024]
                         const float* __restrict__ w,      // [64][64][1024]
                         float* __restrict__ out_c,        // [16][64][1024]
                         int accumulate) {
  const int p = blockIdx.x;
  const int tid = threadIdx.x;
  const int wv = tid >> 5, lane = tid & 31;
  const int lo = lane & 15, hi = lane >> 4;
  const int ob = wv * 16;
  v8f c = zero8();
  if (accumulate) {
#pragma unroll
    for (int r = 0; r < 8; ++r)
      c[r] = out_c[((size_t)((r + 8 * hi) * 64 + ob + lo)) * 1024 + p];
  }
  for (int kk = 0; kk < 64; kk += 4) {
    v2f a, b;
    const int i0 = kk + 2 * hi;
    a.x = in_c[((size_t)(lo * 64 + i0)) * 1024 + p];
    a.y = in_c[((size_t)(lo * 64 + i0 + 1)) * 1024 + p];
    b.x = w[((size_t)(i0 * 64 + ob + lo)) * 1024 + p];
    b.y = w[((size_t)((i0 + 1) * 64 + ob + lo)) * 1024 + p];
    c = wmma4(a, b, c);
  }
#pragma unroll
  for (int r = 0; r < 8; ++r)
    out_c[((size_t)((r + 8 * hi) * 64 + ob + lo)) * 1024 + p] = c[r];
}

extern "C" void kernel_launch(void* const* d_in, const int* in_sizes, int n_in,
                              void* d_out, int out_size, void* d_ws,
                              size_t ws_size, hipStream_t stream) {
  const float* x  = (const float*)d_in[0];
  const float* w1 = (const float*)d_in[1];
  const float* wr = (const float*)d_in[2];
  float* out = (float*)d_out;
  float* ws  = (float*)d_ws;

  float* M1a = ws;                 // 32768
  float* W2a = ws + 32768;         // 32768
  float* M1s = ws + 65536;         // 32768
  float* W2s = ws + 98304;         // 32768
  float* cX  = ws + 131072;        // 1048576 : x_ht corner   [16][64][1024]
  float* cR  = cX + 1048576;       // 1048576 : refine corner
  float* acc = cR + 1048576;       // 1048576 : out_ht corner (accumulated)

  const size_t lds = 128 * 256 * sizeof(float);  // 128 KB dynamic LDS

  init_mats<<<512, 256, 0, stream>>>(M1a, W2a, M1s, W2s);
  // 1) corner DHT of x
  dht_analysis<false><<<NIMG, 256, lds, stream>>>(x, nullptr, M1a, W2a, cX);
  // 2) first spectral channel mix
  chan_mix<<<CORN, 128, 0, stream>>>(cX, w1, acc, 0);
  // 3) inverse DHT of zero-padded spectrum -> x_reconstructed (staged in d_out)
  dht_synthesis<<<NIMG, 256, lds, stream>>>(acc, M1s, W2s, out, 1.0f / 65536.0f);
  // 4+5) mask fused into corner DHT of x_refine
  dht_analysis<true><<<NIMG, 256, lds, stream>>>(x, out, M1a, W2a, cR);
  // 6) refine channel mix, accumulated into out_ht corner
  chan_mix<<<CORN, 128, 0, stream>>>(cR, wr, acc, 1);
  // 7) final inverse DHT -> d_out
  dht_synthesis<<<NIMG, 256, lds, stream>>>(acc, M1s, W2s, out, 1.0f / 65536.0f);
}